// LSTMcell_53601191854843
// MI455X (gfx1250) — compile-verified
//
#include <hip/hip_runtime.h>
#include <hip/hip_bf16.h>
#include <math.h>

// ---------------- types ----------------
typedef __attribute__((ext_vector_type(16))) __bf16        v16bf;
typedef __attribute__((ext_vector_type(8)))  float         v8f;
typedef __attribute__((ext_vector_type(4)))  unsigned int  v4u;

#define HS 2048   // HIDDEN_SIZE
#define BS 4096   // BATCH
#define KD 2048   // K depth (INPUT_SIZE == HIDDEN_SIZE)

union Frag16 { v4u q[2]; v16bf v; };

__device__ __forceinline__ unsigned short f2bf(float f) {
    unsigned u = __float_as_uint(f);
    unsigned r = u + 0x7FFFu + ((u >> 16) & 1u);   // round-to-nearest-even
    return (unsigned short)(r >> 16);
}

// WMMA with C/D hard-tied to the same register tuple ("+v") so accumulation is
// in place and no v_mov_b64 shuffling can be generated. Syntax matches the
// disassembler: v_wmma_f32_16x16x32_bf16 vD[0:7], vA[0:7], vB[0:7], vC[0:7].
__device__ __forceinline__ void wmma_acc(v8f& acc, v16bf a, v16bf b) {
    asm("v_wmma_f32_16x16x32_bf16 %0, %1, %2, %0"
        : "+v"(acc)
        : "v"(a), "v"(b));
}

// WMMA(bf16) -> VALU hazard: 4 coexec NOPs required before VALU touches D.
__device__ __forceinline__ void wmma_hazard_fence() {
    asm volatile("v_nop\n\tv_nop\n\tv_nop\n\tv_nop\n\tv_nop\n\tv_nop" ::: );
}

__device__ __forceinline__ float sigmoidf_(float x) {
    return 1.0f / (1.0f + __expf(-x));
}

// gfx1250 async global->LDS copy, 16B per lane, GVS addressing:
// mem_addr = SGPR64 base + VGPR32 byte offset. VDST = LDS byte address VGPR.
// Tracked by ASYNCcnt.
__device__ __forceinline__ void async_copy_b128(const void* sgpr_base,
                                                unsigned goff_bytes,
                                                unsigned lds_off) {
    asm volatile("global_load_async_to_lds_b128 %0, %1, %2"
                 :: "v"(lds_off), "v"(goff_bytes), "s"(sgpr_base)
                 : "memory");
}

__device__ __forceinline__ void wait_async0() {
    asm volatile("s_wait_asynccnt 0" ::: "memory");
}

// ---------------- pre-pass: f32 -> bf16 (same layout), vectorized x4 ----------------
__global__ void f32_to_bf16_kernel(const float4* __restrict__ src,
                                   uint2* __restrict__ dst, int n4) {
    int i = blockIdx.x * blockDim.x + threadIdx.x;
    if (i >= n4) return;
    float4 v = src[i];
    uint2 o;
    o.x = (unsigned)f2bf(v.x) | ((unsigned)f2bf(v.y) << 16);
    o.y = (unsigned)f2bf(v.z) | ((unsigned)f2bf(v.w) << 16);
    dst[i] = o;
}

// ---------------- pre-pass: state [K=2048][B=4096] f32 -> bf16 [B][K] ----------------
__global__ void transpose_f32_to_bf16_kernel(const float* __restrict__ src,
                                             unsigned short* __restrict__ dst) {
    __shared__ float tile[32][33];
    const int kb = blockIdx.x * 32;   // k tile origin
    const int bb = blockIdx.y * 32;   // b tile origin
    const int tx = threadIdx.x;       // 0..31
    const int ty = threadIdx.y;       // 0..7
#pragma unroll
    for (int r = 0; r < 32; r += 8)
        tile[ty + r][tx] = src[(size_t)(kb + ty + r) * BS + (bb + tx)];
    __syncthreads();
#pragma unroll
    for (int r = 0; r < 32; r += 8)
        dst[(size_t)(bb + ty + r) * KD + (kb + tx)] = f2bf(tile[tx][ty + r]);
}

// ---------------- fused LSTM cell: dual GEMM (bf16 WMMA) + gates ----------------
// Grid: (HS/64, BS/64). Block: 256 threads = 8 waves arranged 4(h) x 2(b).
// Block tile: 64 h-rows (per gate, x4 gates) x 64 batch cols; K step = 32.
// Double-buffered LDS staging via global_load_async_to_lds_b128 (ASYNCcnt).
// Per K step per wave: 8 wmma (4 gates x 2 N-subtiles) fed by 12 ds_load_b128.

#define SA_ELEMS (256 * 32)          // 4 gates x 64 rows x 32 k (bf16)
#define SB_ELEMS (64 * 32)           // 64 batch cols x 32 k (bf16)
#define BUF_ELEMS (SA_ELEMS + SB_ELEMS)      // 10240 shorts = 20 KB
#define NSTEPS 128                   // 2 sources x (2048/32)

__global__ __launch_bounds__(256)
void lstm_wmma_kernel(const unsigned short* __restrict__ Wi,   // [4H][K] bf16
                      const unsigned short* __restrict__ Wh,   // [4H][K] bf16
                      const unsigned short* __restrict__ Xb,   // inp  [B][K] bf16
                      const unsigned short* __restrict__ St,   // state^T [B][K] bf16
                      const float* __restrict__ Bi,            // [4H]
                      const float* __restrict__ Bh,            // [4H]
                      const float* __restrict__ Cin,           // [H][B]
                      float* __restrict__ Out)                 // c1 [H][B] ++ h1 [H][B]
{
    __shared__ __align__(16) unsigned short smem[2][BUF_ELEMS];

    const int t      = threadIdx.x;
    const int lane   = t & 31;
    const int wid    = t >> 5;
    const int wave_h = wid & 3;          // 0..3 -> 16-row slab within 64-row tile
    const int wave_b = wid >> 2;         // 0..1 -> 32-col slab within 64-col tile
    const int khalf  = lane >> 4;        // lane group selects K-half of fragment
    const int l15    = lane & 15;

    const int h0 = blockIdx.x * 64;
    const int b0 = blockIdx.y * 64;

    // LDS byte address base (generic LDS pointer low 32 bits == LDS offset)
    const unsigned sbase = (unsigned)(size_t)&smem[0][0];

    // ---- per-thread staging geometry ----
    // A chunk j covers elements e = (j*256+t)*8 of the staged [256 rows][32 k] tile.
    auto goffA = [&](int j) -> unsigned {
        const int e    = (j * 256 + t) * 8;
        const int row  = e >> 5;          // 0..255 = gate*64 + r
        const int kk   = e & 31;
        const int gate = row >> 6;
        const int r    = row & 63;
        return (unsigned)((gate * HS + h0 + r) * KD + kk) * 2u;   // byte offset
    };
    auto goffB = [&]() -> unsigned {
        return (unsigned)((b0 + (t >> 2)) * KD + (t & 3) * 8) * 2u;
    };
    const unsigned la0 = (unsigned)((0 * 256 + t) * 8) * 2;   // LDS byte offsets
    const unsigned la1 = (unsigned)((1 * 256 + t) * 8) * 2;
    const unsigned la2 = (unsigned)((2 * 256 + t) * 8) * 2;
    const unsigned la3 = (unsigned)((3 * 256 + t) * 8) * 2;
    const unsigned lb  = (unsigned)(t * 8) * 2 + SA_ELEMS * 2;

    // Loop-carried 32-bit global byte offsets (advance 64 B per step); uniform
    // SGPR base pointers select the matrix.
    const void* aB = Wi;
    const void* bB = Xb;
    unsigned o0 = goffA(0), o1 = goffA(1), o2 = goffA(2), o3 = goffA(3);
    unsigned ob = goffB();

    // Accumulators (a<gate><n-subtile>); asm "+v" ties C==D in place.
    v8f a00 = {}, a01 = {}, a10 = {}, a11 = {};
    v8f a20 = {}, a21 = {}, a30 = {}, a31 = {};

    // Prologue: stage step 0 into buffer 0.
    async_copy_b128(aB, o0, sbase + la0); o0 += 64;
    async_copy_b128(aB, o1, sbase + la1); o1 += 64;
    async_copy_b128(aB, o2, sbase + la2); o2 += 64;
    async_copy_b128(aB, o3, sbase + la3); o3 += 64;
    async_copy_b128(bB, ob, sbase + lb);  ob += 64;

    for (int it = 0; it < NSTEPS; ++it) {
        wait_async0();        // my async copies for smem[it&1] complete
        __syncthreads();      // everyone's copies visible; prev-buf reads done

        const int st = it + 1;
        if (st < NSTEPS) {    // overlap next-stage DMA with this step's WMMAs
            if (st == 64) {   // source switch: input GEMM -> hidden GEMM
                aB = Wh; bB = St;
                o0 = goffA(0); o1 = goffA(1); o2 = goffA(2); o3 = goffA(3);
                ob = goffB();
            }
            const unsigned ab = sbase + (unsigned)(st & 1) * (BUF_ELEMS * 2);
            async_copy_b128(aB, o0, ab + la0); o0 += 64;
            async_copy_b128(aB, o1, ab + la1); o1 += 64;
            async_copy_b128(aB, o2, ab + la2); o2 += 64;
            async_copy_b128(aB, o3, ab + la3); o3 += 64;
            async_copy_b128(bB, ob, ab + lb);  ob += 64;
        }

        const unsigned short* sA = &smem[it & 1][0];
        const unsigned short* sB = sA + SA_ELEMS;

        // B fragments (N x K rows): lanes 0-15 K 0..15, lanes 16-31 K 16..31.
        Frag16 bf0, bf1;
        {
            const int e0 = (wave_b * 32 + l15) * 32 + khalf * 16;
            bf0.q[0] = *(const v4u*)&sB[e0];
            bf0.q[1] = *(const v4u*)&sB[e0 + 8];
            const int e1 = (wave_b * 32 + 16 + l15) * 32 + khalf * 16;
            bf1.q[0] = *(const v4u*)&sB[e1];
            bf1.q[1] = *(const v4u*)&sB[e1 + 8];
        }
        // A fragments per gate: lanes 0-15 K {0..7,16..23}, lanes 16-31 {8..15,24..31}.
        const int ea = (wave_h * 16 + l15) * 32 + khalf * 8;
        Frag16 af;
        af.q[0] = *(const v4u*)&sA[ea];
        af.q[1] = *(const v4u*)&sA[ea + 16];
        wmma_acc(a00, af.v, bf0.v);
        wmma_acc(a01, af.v, bf1.v);
        af.q[0] = *(const v4u*)&sA[64 * 32 + ea];
        af.q[1] = *(const v4u*)&sA[64 * 32 + ea + 16];
        wmma_acc(a10, af.v, bf0.v);
        wmma_acc(a11, af.v, bf1.v);
        af.q[0] = *(const v4u*)&sA[128 * 32 + ea];
        af.q[1] = *(const v4u*)&sA[128 * 32 + ea + 16];
        wmma_acc(a20, af.v, bf0.v);
        wmma_acc(a21, af.v, bf1.v);
        af.q[0] = *(const v4u*)&sA[192 * 32 + ea];
        af.q[1] = *(const v4u*)&sA[192 * 32 + ea + 16];
        wmma_acc(a30, af.v, bf0.v);
        wmma_acc(a31, af.v, bf1.v);
    }

    // Manual WMMA->VALU hazard fence before the epilogue reads accumulators.
    wmma_hazard_fence();

    // Epilogue: C/D layout -> VGPR j holds M = j + 8*khalf, N = l15.
    auto epi = [&](v8f Ai, v8f Af, v8f Ag, v8f Ao, int b) {
#pragma unroll
        for (int j = 0; j < 8; ++j) {
            const int h = h0 + wave_h * 16 + khalf * 8 + j;
            const float xi = Ai[j] + Bi[0 * HS + h] + Bh[0 * HS + h];
            const float xf = Af[j] + Bi[1 * HS + h] + Bh[1 * HS + h];
            const float xg = Ag[j] + Bi[2 * HS + h] + Bh[2 * HS + h];
            const float xo = Ao[j] + Bi[3 * HS + h] + Bh[3 * HS + h];
            const float ig = sigmoidf_(xi);
            const float fg = sigmoidf_(xf);
            const float gg = tanhf(xg);
            const float og = sigmoidf_(xo);
            const float cprev = Cin[(size_t)h * BS + b];
            const float c1 = fg * cprev + ig * gg;
            const float h1 = og * tanhf(c1);
            Out[(size_t)h * BS + b] = c1;
            Out[(size_t)HS * BS + (size_t)h * BS + b] = h1;
        }
    };
    epi(a00, a10, a20, a30, b0 + wave_b * 32 + l15);
    epi(a01, a11, a21, a31, b0 + wave_b * 32 + 16 + l15);
}

// ---------------- host glue ----------------
extern "C" void kernel_launch(void* const* d_in, const int* in_sizes, int n_in,
                              void* d_out, int out_size, void* d_ws, size_t ws_size,
                              hipStream_t stream) {
    const float* inp   = (const float*)d_in[0];   // [4096][2048]
    const float* state = (const float*)d_in[1];   // [2048][4096]
    const float* c     = (const float*)d_in[2];   // [2048][4096]
    const float* Wi    = (const float*)d_in[3];   // [8192][2048]
    const float* Bi    = (const float*)d_in[4];   // [8192]
    const float* Wh    = (const float*)d_in[5];   // [8192][2048]
    const float* Bh    = (const float*)d_in[6];   // [8192]
    float* out = (float*)d_out;

    // Workspace partition (bf16 copies), total 96 MB.
    char* ws = (char*)d_ws;
    unsigned short* Wi_b = (unsigned short*)(ws);                      // 32 MB
    unsigned short* Wh_b = (unsigned short*)(ws + (size_t)33554432);   // 32 MB
    unsigned short* X_b  = (unsigned short*)(ws + (size_t)67108864);   // 16 MB
    unsigned short* S_b  = (unsigned short*)(ws + (size_t)83886080);   // 16 MB

    // f32 -> bf16 conversions (vectorized by 4).
    {
        const int nW4 = (4 * HS * KD) / 4;   // 4,194,304
        f32_to_bf16_kernel<<<nW4 / 256, 256, 0, stream>>>(
            (const float4*)Wi, (uint2*)Wi_b, nW4);
        f32_to_bf16_kernel<<<nW4 / 256, 256, 0, stream>>>(
            (const float4*)Wh, (uint2*)Wh_b, nW4);
        const int nX4 = (BS * KD) / 4;       // 2,097,152
        f32_to_bf16_kernel<<<nX4 / 256, 256, 0, stream>>>(
            (const float4*)inp, (uint2*)X_b, nX4);
    }
    // state [K][B] -> bf16 [B][K]
    {
        dim3 grid(KD / 32, BS / 32);
        dim3 block(32, 8);
        transpose_f32_to_bf16_kernel<<<grid, block, 0, stream>>>(state, S_b);
    }
    // Fused dual-GEMM + LSTM gates.
    {
        dim3 grid(HS / 64, BS / 64);
        lstm_wmma_kernel<<<grid, 256, 0, stream>>>(Wi_b, Wh_b, X_b, S_b,
                                                   Bi, Bh, c, out);
    }
}